// GolbalContextBlock_29609504538665
// MI455X (gfx1250) — compile-verified
//
#include <hip/hip_runtime.h>
#include <math.h>

#define HEADS   8
#define CDIM    256
#define C1      32
#define HW      4096
#define BATCH   32
#define PLANES  64
#define LN_EPS  1e-3f

typedef __attribute__((ext_vector_type(2))) float v2f;
typedef __attribute__((ext_vector_type(8))) float v8f;

// ---------------------------------------------------------------------------
// Kernel 1: softmax-attention spatial pooling.
// One block per n = b*HEADS + head (256 blocks). Each thread streams 16
// positions (32 channels each) with an online softmax, then LDS tree-merge.
// Reads x exactly once (128 MB total across the grid).
// ---------------------------------------------------------------------------
__global__ __launch_bounds__(256) void gc_context_kernel(
    const float* __restrict__ x, const float* __restrict__ wm,
    const float* __restrict__ bm, float* __restrict__ ctx) {
  __shared__ float s_wm[C1];
  __shared__ float s_m[256];
  __shared__ float s_d[256];
  __shared__ float s_acc[256][C1 + 1];  // +1 pad vs bank conflicts

  const int t = threadIdx.x;
  const int n = blockIdx.x;          // 0..255
  const int b = n >> 3, head = n & 7;

  if (t < C1) s_wm[t] = wm[t];
  __syncthreads();

  float4 wv[8];
#pragma unroll
  for (int i = 0; i < 8; ++i) wv[i] = ((const float4*)s_wm)[i];
  const float bias = bm[0];

  const float* base = x + (size_t)b * HW * CDIM + (size_t)head * C1;

  float m = -3.0e38f, d = 0.0f;
  float4 acc[8];
#pragma unroll
  for (int i = 0; i < 8; ++i) acc[i] = make_float4(0.f, 0.f, 0.f, 0.f);

  for (int s = t; s < HW; s += 256) {
    const float4* p = (const float4*)(base + (size_t)s * CDIM);
    float4 v[8];
#pragma unroll
    for (int i = 0; i < 8; ++i) v[i] = p[i];

    float logit = bias;
#pragma unroll
    for (int i = 0; i < 8; ++i)
      logit += v[i].x * wv[i].x + v[i].y * wv[i].y +
               v[i].z * wv[i].z + v[i].w * wv[i].w;

    if (logit > m) {
      const float sc = __expf(m - logit);
      d *= sc;
#pragma unroll
      for (int i = 0; i < 8; ++i) {
        acc[i].x *= sc; acc[i].y *= sc; acc[i].z *= sc; acc[i].w *= sc;
      }
      m = logit;
    }
    const float w = __expf(logit - m);
    d += w;
#pragma unroll
    for (int i = 0; i < 8; ++i) {
      acc[i].x += w * v[i].x; acc[i].y += w * v[i].y;
      acc[i].z += w * v[i].z; acc[i].w += w * v[i].w;
    }
  }

  s_m[t] = m;
  s_d[t] = d;
#pragma unroll
  for (int i = 0; i < 8; ++i) {
    s_acc[t][4 * i + 0] = acc[i].x; s_acc[t][4 * i + 1] = acc[i].y;
    s_acc[t][4 * i + 2] = acc[i].z; s_acc[t][4 * i + 3] = acc[i].w;
  }
  __syncthreads();

  // online-softmax merge tree
  for (int stride = 128; stride >= 1; stride >>= 1) {
    if (t < stride) {
      const float m1 = s_m[t], m2 = s_m[t + stride];
      const float M = fmaxf(m1, m2);
      const float sc1 = __expf(m1 - M), sc2 = __expf(m2 - M);
      s_d[t] = s_d[t] * sc1 + s_d[t + stride] * sc2;
      s_m[t] = M;
      for (int c = 0; c < C1; ++c)
        s_acc[t][c] = s_acc[t][c] * sc1 + s_acc[t + stride][c] * sc2;
    }
    __syncthreads();
  }

  if (t < C1) ctx[n * C1 + t] = s_acc[0][t] / s_d[0];
}

// ---------------------------------------------------------------------------
// Kernel 2: bottleneck MLP on ctx (32x256) using V_WMMA_F32_16X16X4_F32.
//   y = LN(ctx @ w1 + b1) -> ReLU ;  term = y @ w2 + b2
// Single block, 8 waves. fp32 WMMA keeps reference precision.
// Lane layouts per CDNA5 ISA 7.12.2:
//   A 16x4 : lane L, vgpr v -> (M = L%16, K = 2*(L/16)+v)
//   B 4x16 : lane L, vgpr v -> (K = 2*(L/16)+v, N = L%16)
//   D 16x16: lane L, vgpr v -> (M = v + 8*(L/16), N = L%16)
// ---------------------------------------------------------------------------
__global__ __launch_bounds__(256) void gc_mlp_kernel(
    const float* __restrict__ ctx, const float* __restrict__ w1,
    const float* __restrict__ b1, const float* __restrict__ gamma,
    const float* __restrict__ beta, const float* __restrict__ w2,
    const float* __restrict__ b2, float* __restrict__ term) {
  __shared__ float s_y[BATCH][PLANES];

  const int t = threadIdx.x;
  const int wave = t >> 5;
  const int lane = t & 31;
  const int lrow = lane & 15;
  const int lhalf = lane >> 4;

  // ---- GEMM1: (32x256)@(256x64), 8 tiles of 16x16, one per wave ----
  {
    const int mt = wave >> 2;  // 0..1
    const int nt = wave & 3;   // 0..3
    const int arow = mt * 16 + lrow;
    const int bcol = nt * 16 + lrow;
    v8f c = {};
    for (int k0 = 0; k0 < CDIM; k0 += 4) {
      const int ka = k0 + lhalf * 2;
      v2f a, bv;
      a.x = ctx[arow * CDIM + ka];
      a.y = ctx[arow * CDIM + ka + 1];
      bv.x = w1[ka * PLANES + bcol];
      bv.y = w1[(ka + 1) * PLANES + bcol];
      c = __builtin_amdgcn_wmma_f32_16x16x4_f32(false, a, false, bv,
                                                (short)0, c, false, false);
    }
    const float bb = b1[bcol];
#pragma unroll
    for (int v = 0; v < 8; ++v) {
      const int row = mt * 16 + v + 8 * lhalf;
      s_y[row][bcol] = c[v] + bb;
    }
  }
  __syncthreads();

  // ---- LayerNorm over planes + ReLU (one thread per batch row) ----
  if (t < BATCH) {
    float mu = 0.f;
    for (int c = 0; c < PLANES; ++c) mu += s_y[t][c];
    mu *= (1.0f / PLANES);
    float var = 0.f;
    for (int c = 0; c < PLANES; ++c) {
      const float dd = s_y[t][c] - mu;
      var += dd * dd;
    }
    var *= (1.0f / PLANES);
    const float rs = rsqrtf(var + LN_EPS);
    for (int c = 0; c < PLANES; ++c) {
      const float yv = (s_y[t][c] - mu) * rs * gamma[c] + beta[c];
      s_y[t][c] = fmaxf(yv, 0.0f);
    }
  }
  __syncthreads();

  // ---- GEMM2: (32x64)@(64x256), 32 tiles, 4 per wave (uniform -> EXEC=all1) --
  for (int tile = wave; tile < 32; tile += 8) {
    const int mt = tile >> 4;  // 0..1
    const int nt = tile & 15;  // 0..15
    const int arow = mt * 16 + lrow;
    const int bcol = nt * 16 + lrow;
    v8f c = {};
    for (int k0 = 0; k0 < PLANES; k0 += 4) {
      const int ka = k0 + lhalf * 2;
      v2f a, bv;
      a.x = s_y[arow][ka];
      a.y = s_y[arow][ka + 1];
      bv.x = w2[ka * CDIM + bcol];
      bv.y = w2[(ka + 1) * CDIM + bcol];
      c = __builtin_amdgcn_wmma_f32_16x16x4_f32(false, a, false, bv,
                                                (short)0, c, false, false);
    }
    const float bb = b2[bcol];
#pragma unroll
    for (int v = 0; v < 8; ++v) {
      const int row = mt * 16 + v + 8 * lhalf;
      term[row * CDIM + bcol] = c[v] + bb;
    }
  }
}

// ---------------------------------------------------------------------------
// Kernel 3: out = x + term[b, :] broadcast. 4096 blocks x 256 threads,
// each block owns 2048 contiguous float4 (one batch sample per block),
// term row cached in LDS. Pure streaming float4 traffic.
// ---------------------------------------------------------------------------
__global__ __launch_bounds__(256) void gc_add_kernel(
    const float* __restrict__ x, const float* __restrict__ term,
    float* __restrict__ out) {
  __shared__ float4 s_t4[CDIM / 4];

  const size_t chunk = (size_t)blockIdx.x * 2048;  // float4 units
  const int b = (int)(chunk >> 18);                // 262144 float4 per sample
  const int t = threadIdx.x;

  if (t < CDIM / 4) s_t4[t] = ((const float4*)term)[b * (CDIM / 4) + t];
  __syncthreads();

  const float4* xp = (const float4*)x + chunk;
  float4* op = (float4*)out + chunk;
#pragma unroll
  for (int k = 0; k < 8; ++k) {
    const int j = k * 256 + t;
    float4 xv = xp[j];
    const float4 tv = s_t4[j & 63];
    xv.x += tv.x; xv.y += tv.y; xv.z += tv.z; xv.w += tv.w;
    op[j] = xv;
  }
}

extern "C" void kernel_launch(void* const* d_in, const int* in_sizes, int n_in,
                              void* d_out, int out_size, void* d_ws,
                              size_t ws_size, hipStream_t stream) {
  (void)in_sizes; (void)n_in; (void)out_size; (void)ws_size;
  const float* x     = (const float*)d_in[0];
  const float* wm    = (const float*)d_in[1];
  const float* bm    = (const float*)d_in[2];
  const float* w1    = (const float*)d_in[3];
  const float* b1    = (const float*)d_in[4];
  const float* gamma = (const float*)d_in[5];
  const float* beta  = (const float*)d_in[6];
  const float* w2    = (const float*)d_in[7];
  const float* b2    = (const float*)d_in[8];
  float* out = (float*)d_out;

  float* ctx  = (float*)d_ws;          // 256*32  = 8192 floats
  float* term = ctx + BATCH * HEADS * C1;  // 32*256 = 8192 floats

  gc_context_kernel<<<BATCH * HEADS, 256, 0, stream>>>(x, wm, bm, ctx);
  gc_mlp_kernel<<<1, 256, 0, stream>>>(ctx, w1, b1, gamma, beta, w2, b2, term);
  gc_add_kernel<<<4096, 256, 0, stream>>>(x, term, out);
}